// TargetAttention_38062000177789
// MI455X (gfx1250) — compile-verified
//
#include <hip/hip_runtime.h>

// Problem constants (from reference)
#define B_ 8
#define D_ 512
#define L_ 2500
#define N_ 8921

typedef __attribute__((ext_vector_type(2))) float v2f;
typedef __attribute__((ext_vector_type(8))) float v8f;

#define NT    16    // n-tile (WMMA M)
#define KC    32    // K chunk over D for GEMM1
#define LBLK  512   // l-tile per block for GEMM1 (8 waves x 64)
#define LPADP 516   // H-tile pair-row pitch (bank shift 8/row, 16B multiple)
#define KC2   32    // K chunk over L for GEMM2
#define QPAD  36    // 16-row / 512-row tile pitch (conflict-free, 16B multiple)

#define WMMA_F32(acc, a, b) \
  __builtin_amdgcn_wmma_f32_16x16x4_f32(false, (a), false, (b), (short)0, (acc), false, false)

// ---------------------------------------------------------------------------
// Kernel 1: E[b,n,l] = sum_d Qw[n,d] * H[b,d,l]      (logits, written to A)
// grid: (ceil(N/16), ceil(L/512), B), block: 256 (8 waves, 4 acc tiles each)
// H tile stored K-pair-interleaved: Hsp[kp][l][0..1] = H[2kp..2kp+1][l]
// -> each B-fragment is one aligned ds_load_b64 (no repacking movs).
// ---------------------------------------------------------------------------
__global__ __launch_bounds__(256) void ta_gemm1_logits(
    const float* __restrict__ Qw, const float* __restrict__ H,
    float* __restrict__ E)
{
  __shared__ float Qs[NT][QPAD];           // 16 x 36
  __shared__ float Hsp[KC / 2][LPADP][2];  // 16 x 516 x 2

  const int tid  = threadIdx.x;
  const int wave = tid >> 5;
  const int lane = tid & 31;
  const int n0 = blockIdx.x * NT;
  const int l0 = blockIdx.y * LBLK;
  const int b  = blockIdx.z;

  const int row16 = lane & 15;          // A-matrix row M / B-matrix column
  const int kofs  = (lane >> 4) << 1;   // K offset per half-wave (0 or 2)
  const int khalf = lane >> 4;          // which element of the K-quad pair
  const int lbase = wave * 64;          // this wave's 64-wide l slice

  v8f acc0 = {0.f,0.f,0.f,0.f,0.f,0.f,0.f,0.f};
  v8f acc1 = acc0, acc2 = acc0, acc3 = acc0;

  const float* Hb = H + (size_t)b * D_ * L_;

  for (int k0 = 0; k0 < D_; k0 += KC) {
    // Stage Qw tile 16x32 (128 float4s, threads 0..127)
    if (tid < 128) {
      const int r  = tid >> 3;         // 0..15
      const int c4 = (tid & 7) << 2;   // 0,4,...,28
      const int n  = n0 + r;
      float4 v = make_float4(0.f, 0.f, 0.f, 0.f);
      if (n < N_) v = *(const float4*)&Qw[(size_t)n * D_ + (k0 + c4)];
      *(float4*)&Qs[r][c4] = v;
    }
    // Stage H tile 32x512 as 16 K-pair rows (interleave two rows per store)
    {
      const int c4 = (tid & 127) << 2; // l offset 0..508 (multiple of 4)
      const int rb = tid >> 7;         // 0..1
      const int l  = l0 + c4;
      const bool inL = (l < L_);       // L_%4==0: float4 never straddles
#pragma unroll
      for (int i = 0; i < 8; ++i) {
        const int kp = rb + i * 2;     // 0..15
        float4 va = make_float4(0.f, 0.f, 0.f, 0.f);
        float4 vb = va;
        if (inL) {
          va = *(const float4*)&Hb[(size_t)(k0 + 2 * kp    ) * L_ + l];
          vb = *(const float4*)&Hb[(size_t)(k0 + 2 * kp + 1) * L_ + l];
        }
        const float4 p0 = make_float4(va.x, vb.x, va.y, vb.y);
        const float4 p1 = make_float4(va.z, vb.z, va.w, vb.w);
        *(float4*)&Hsp[kp][c4    ][0] = p0;   // 16B aligned: (kp*516+c4) even
        *(float4*)&Hsp[kp][c4 + 2][0] = p1;
      }
    }
    __syncthreads();

#pragma unroll
    for (int kk = 0; kk < KC / 4; ++kk) {
      const int kb = kk * 4 + kofs;
      const int kp = kk * 2 + khalf;   // pair row holding (kb, kb+1)
      const v2f a  = *(const v2f*)&Qs[row16][kb];
      const v2f b0 = *(const v2f*)&Hsp[kp][lbase +  0 + row16][0];
      const v2f b1 = *(const v2f*)&Hsp[kp][lbase + 16 + row16][0];
      const v2f b2 = *(const v2f*)&Hsp[kp][lbase + 32 + row16][0];
      const v2f b3 = *(const v2f*)&Hsp[kp][lbase + 48 + row16][0];
      acc0 = WMMA_F32(acc0, a, b0);
      acc1 = WMMA_F32(acc1, a, b1);
      acc2 = WMMA_F32(acc2, a, b2);
      acc3 = WMMA_F32(acc3, a, b3);
    }
    __syncthreads();
  }

  // Store: VGPR i -> row i (lanes 0-15) / row i+8 (lanes 16-31)
  const int rhalf = (lane >> 4) * 8;
  float* Eb = E + (size_t)b * N_ * L_;
  v8f* accs[4] = {&acc0, &acc1, &acc2, &acc3};
#pragma unroll
  for (int s = 0; s < 4; ++s) {
    const int l = l0 + lbase + s * 16 + row16;
    if (l < L_) {
#pragma unroll
      for (int i = 0; i < 8; ++i) {
        const int n = n0 + rhalf + i;
        if (n < N_) Eb[(size_t)n * L_ + l] = (*accs[s])[i];
      }
    }
  }
}

// ---------------------------------------------------------------------------
// Kernel 2: in-place row softmax over L. grid: B*N blocks, block: 256.
// Row cached in registers (ceil(2500/256) = 10 per thread): 1 read + 1 write.
// ---------------------------------------------------------------------------
__global__ __launch_bounds__(256) void ta_softmax_rows(float* __restrict__ A)
{
  __shared__ float red[256];
  const int tid = threadIdx.x;
  float* p = A + (size_t)blockIdx.x * L_;

  float vals[10];
  int cnt = 0;
  float m = -3.402823466e38f;
  for (int l = tid; l < L_; l += 256) {
    const float v = p[l];
    vals[cnt++] = v;
    m = fmaxf(m, v);
  }
  red[tid] = m;
  __syncthreads();
#pragma unroll
  for (int s = 128; s > 0; s >>= 1) {
    if (tid < s) red[tid] = fmaxf(red[tid], red[tid + s]);
    __syncthreads();
  }
  m = red[0];
  __syncthreads();

  float sum = 0.f;
  for (int i = 0; i < cnt; ++i) {
    vals[i] = __expf(vals[i] - m);
    sum += vals[i];
  }
  red[tid] = sum;
  __syncthreads();
#pragma unroll
  for (int s = 128; s > 0; s >>= 1) {
    if (tid < s) red[tid] += red[tid + s];
    __syncthreads();
  }
  const float inv = 1.f / red[0];

  cnt = 0;
  for (int l = tid; l < L_; l += 256) p[l] = vals[cnt++] * inv;
}

// ---------------------------------------------------------------------------
// Kernel 3: C[b,n,d] = sum_l A[b,n,l] * H[b,d,l]
// grid: (ceil(N/16), B), block: 256 (8 waves x 64 d = all of D per block)
// Fragments are adjacent (row-major, even K offsets) -> ds_load_b64 each.
// ---------------------------------------------------------------------------
__global__ __launch_bounds__(256) void ta_gemm2_context(
    const float* __restrict__ A, const float* __restrict__ H,
    float* __restrict__ C)
{
  __shared__ float As[NT][QPAD];      // 16 x 36
  __shared__ float Hs2[D_][QPAD];     // 512 x 36

  const int tid  = threadIdx.x;
  const int wave = tid >> 5;
  const int lane = tid & 31;
  const int n0 = blockIdx.x * NT;
  const int b  = blockIdx.y;

  const int row16 = lane & 15;
  const int kofs  = (lane >> 4) << 1;
  const int dbase = wave * 64;

  const float* Ab = A + (size_t)b * N_ * L_;
  const float* Hb = H + (size_t)b * D_ * L_;

  v8f acc0 = {0.f,0.f,0.f,0.f,0.f,0.f,0.f,0.f};
  v8f acc1 = acc0, acc2 = acc0, acc3 = acc0;

  for (int k0 = 0; k0 < L_; k0 += KC2) {
    // Stage A tile 16x32 (128 float4s, threads 0..127); zero-fill OOB
    if (tid < 128) {
      const int r  = tid >> 3;
      const int c4 = (tid & 7) << 2;
      const int n  = n0 + r;
      const int l  = k0 + c4;
      float4 v = make_float4(0.f, 0.f, 0.f, 0.f);
      if (n < N_ && l < L_) {
        v = *(const float4*)&Ab[(size_t)n * L_ + l];
        if (l + KC2 < L_) __builtin_prefetch(&Ab[(size_t)n * L_ + l + KC2], 0, 0);
      }
      *(float4*)&As[r][c4] = v;
    }
    // Stage H tile 512x32 (4096 float4s, 16 per thread, coalesced over l)
    {
      const int c4 = (tid & 7) << 2;   // 0..28
      const int rb = tid >> 3;         // 0..31
      const int l  = k0 + c4;
      const bool inL = (l < L_);
#pragma unroll
      for (int i = 0; i < 16; ++i) {
        const int r = rb + i * 32;     // 0..511
        float4 v = make_float4(0.f, 0.f, 0.f, 0.f);
        if (inL) {
          v = *(const float4*)&Hb[(size_t)r * L_ + l];
          if (i == 0 && l + KC2 < L_)
            __builtin_prefetch(&Hb[(size_t)r * L_ + l + KC2], 0, 0);
        }
        *(float4*)&Hs2[r][c4] = v;
      }
    }
    __syncthreads();

#pragma unroll
    for (int kk = 0; kk < KC2 / 4; ++kk) {
      const int kb = kk * 4 + kofs;
      const v2f a  = *(const v2f*)&As[row16][kb];              // rows = n, K = l
      const v2f b0 = *(const v2f*)&Hs2[dbase +  0 + row16][kb]; // cols = d
      const v2f b1 = *(const v2f*)&Hs2[dbase + 16 + row16][kb];
      const v2f b2 = *(const v2f*)&Hs2[dbase + 32 + row16][kb];
      const v2f b3 = *(const v2f*)&Hs2[dbase + 48 + row16][kb];
      acc0 = WMMA_F32(acc0, a, b0);
      acc1 = WMMA_F32(acc1, a, b1);
      acc2 = WMMA_F32(acc2, a, b2);
      acc3 = WMMA_F32(acc3, a, b3);
    }
    __syncthreads();
  }

  const int rhalf = (lane >> 4) * 8;
  float* Cb = C + (size_t)b * N_ * D_;
  v8f* accs[4] = {&acc0, &acc1, &acc2, &acc3};
#pragma unroll
  for (int s = 0; s < 4; ++s) {
    const int d = dbase + s * 16 + row16;   // always < 512
#pragma unroll
    for (int i = 0; i < 8; ++i) {
      const int n = n0 + rhalf + i;
      if (n < N_) Cb[(size_t)n * D_ + d] = (*accs[s])[i];
    }
  }
}

// ---------------------------------------------------------------------------
extern "C" void kernel_launch(void* const* d_in, const int* in_sizes, int n_in,
                              void* d_out, int out_size, void* d_ws, size_t ws_size,
                              hipStream_t stream) {
  const float* H  = (const float*)d_in[0];   // [B, D, L]
  const float* Qw = (const float*)d_in[1];   // [N, D]

  float* C = (float*)d_out;                                 // [B, N, D]
  float* A = (float*)d_out + (size_t)B_ * N_ * D_;          // [B, N, L]

  dim3 g1((N_ + NT - 1) / NT, (L_ + LBLK - 1) / LBLK, B_);
  ta_gemm1_logits<<<g1, 256, 0, stream>>>(Qw, H, A);

  ta_softmax_rows<<<B_ * N_, 256, 0, stream>>>(A);

  dim3 g2((N_ + NT - 1) / NT, B_);
  ta_gemm2_context<<<g2, 256, 0, stream>>>(A, H, C);
}